// VBLDS_14989435863663
// MI455X (gfx1250) — compile-verified
//
#include <hip/hip_runtime.h>
#include <math.h>

// ============================================================================
// VBLDS forward (Kalman filter + RTS smoother + E-step stats) for MI455X
// (gfx1250, wave32, WMMA).  All 32x32 f32 matmuls use V_WMMA_F32_16X16X4_F32.
// dimz = 32, dimx = 16 (padded to 32), N = in_sizes[0]/16.
//
// mm32 register-caches all A/B fragments (halves LDS traffic) and interleaves
// the 4 output-tile accumulator chains over k for 4-way WMMA ILP -- a single
// wave has no co-resident wave to hide XDL latency, so ILP must come from
// independent accumulators.
// ============================================================================

typedef __attribute__((ext_vector_type(2))) float v2f;
typedef __attribute__((ext_vector_type(8))) float v8f;

#define MS   33   // LDS row stride (33 floats -> conflict-free column access)
#define ZD   32   // dimz
#define XDIM 16   // dimx

// ---------------------------------------------------------------------------
// Wave-level helpers.  All sequential kernels launch exactly one wave32
// (blockDim.x == 32, all lanes active -> EXEC all ones, required by WMMA).
// Matrices live in LDS as 32 rows x stride MS.
// ---------------------------------------------------------------------------

// D = A @ B (32x32 f32).  Fragments preloaded to VGPRs, 32 WMMAs in 4
// independent accumulator chains interleaved over k.
__device__ __forceinline__ void mm32(const float* A, const float* B, float* D) {
  const int l  = threadIdx.x & 31;
  const int mn = l & 15;            // row (A) / col (B,D) within 16-tile
  const int kh = (l >> 4) << 1;     // K sub-offset per lane half: 0 or 2
  const int vh = (l >> 4) << 3;     // C/D row offset per lane half: 0 or 8

  v2f a[2][8], b[2][8];
#pragma unroll
  for (int k = 0; k < 8; ++k) {
    const int k0 = 4 * k + kh;
    a[0][k].x = A[mn * MS + k0];
    a[0][k].y = A[mn * MS + k0 + 1];
    a[1][k].x = A[(16 + mn) * MS + k0];
    a[1][k].y = A[(16 + mn) * MS + k0 + 1];
    b[0][k].x = B[k0 * MS + mn];
    b[0][k].y = B[(k0 + 1) * MS + mn];
    b[1][k].x = B[k0 * MS + 16 + mn];
    b[1][k].y = B[(k0 + 1) * MS + 16 + mn];
  }
  v8f acc[2][2];
#pragma unroll
  for (int ti = 0; ti < 2; ++ti)
#pragma unroll
    for (int tj = 0; tj < 2; ++tj)
#pragma unroll
      for (int v = 0; v < 8; ++v) acc[ti][tj][v] = 0.0f;
#pragma unroll
  for (int k = 0; k < 8; ++k) {
#pragma unroll
    for (int ti = 0; ti < 2; ++ti)
#pragma unroll
      for (int tj = 0; tj < 2; ++tj)
        acc[ti][tj] = __builtin_amdgcn_wmma_f32_16x16x4_f32(
            /*neg_a=*/false, a[ti][k], /*neg_b=*/false, b[tj][k],
            /*c_mod=*/(short)0, acc[ti][tj], /*reuse_a=*/false,
            /*reuse_b=*/false);
  }
#pragma unroll
  for (int ti = 0; ti < 2; ++ti)
#pragma unroll
    for (int tj = 0; tj < 2; ++tj)
#pragma unroll
      for (int v = 0; v < 8; ++v)
        D[(ti * 16 + vh + v) * MS + (tj * 16 + mn)] = acc[ti][tj][v];
  __syncthreads();
}

__device__ __forceinline__ void addI(float* M) {
  int l = threadIdx.x & 31;
  M[l * MS + l] += 1.0f;
  __syncthreads();
}

__device__ __forceinline__ void negAddI(float* M) {  // M = I - M
  int l = threadIdx.x & 31;
  for (int i = 0; i < ZD; ++i)
    M[i * MS + l] = ((i == l) ? 1.0f : 0.0f) - M[i * MS + l];
  __syncthreads();
}

__device__ __forceinline__ void copyM(const float* S, float* D) {
  int l = threadIdx.x & 31;
  for (int i = 0; i < ZD; ++i) D[i * MS + l] = S[i * MS + l];
  __syncthreads();
}

__device__ __forceinline__ void transM(const float* S, float* D) {
  int l = threadIdx.x & 31;
  for (int i = 0; i < ZD; ++i) D[i * MS + l] = S[l * MS + i];
  __syncthreads();
}

__device__ __forceinline__ void symM(float* M, float* T) {  // M = 0.5(M+M^T)
  int l = threadIdx.x & 31;
  for (int i = 0; i < ZD; ++i)
    T[i * MS + l] = 0.5f * (M[i * MS + l] + M[l * MS + i]);
  __syncthreads();
  for (int i = 0; i < ZD; ++i) M[i * MS + l] = T[i * MS + l];
  __syncthreads();
}

// Gauss-Jordan: Inv = inverse(Mw), Mw destroyed.  Returns log|det|.
__device__ __forceinline__ float inv32(float* Mw, float* Inv) {
  int l = threadIdx.x & 31;
  for (int i = 0; i < ZD; ++i) Inv[i * MS + l] = (i == l) ? 1.0f : 0.0f;
  __syncthreads();
  float ld = 0.0f;
  for (int k = 0; k < ZD; ++k) {
    float p = Mw[k * MS + k];
    ld += logf(fabsf(p));
    float pr = 1.0f / p;
    float mk = Mw[k * MS + l] * pr;
    float ik = Inv[k * MS + l] * pr;
    __syncthreads();
    Mw[k * MS + l] = mk;
    Inv[k * MS + l] = ik;
    __syncthreads();
    for (int i = 0; i < ZD; ++i) {
      if (i == k) continue;
      float f = Mw[i * MS + k];
      Mw[i * MS + l]  -= f * mk;
      Inv[i * MS + l] -= f * ik;
    }
    __syncthreads();
  }
  return ld;
}

// y = M @ v  (lane l owns row l)
__device__ __forceinline__ void mv32(const float* M, const float* v, float* y) {
  int l = threadIdx.x & 31;
  float s = 0.0f;
  for (int j = 0; j < ZD; ++j) s += M[l * MS + j] * v[j];
  __syncthreads();
  y[l] = s;
  __syncthreads();
}

// W = I - M @ inv(I + B@M) @ B   (uses T1,T2,T3 scratch)
__device__ __forceinline__ void woodbury(const float* M, const float* B, float* W,
                                         float* T1, float* T2, float* T3) {
  mm32(B, M, T1);
  addI(T1);
  inv32(T1, T2);
  mm32(M, T2, T3);
  mm32(T3, B, W);
  negAddI(W);
}

// ---------------------------------------------------------------------------
// Kernel 1: forward Kalman filter (sequential in n). One wave32.
// Writes muf[z*N+n] and Vf[(i*32+j)*N+n].
// ---------------------------------------------------------------------------
__global__ __launch_bounds__(32)
void vblds_filter(const float* __restrict__ gx, const float* __restrict__ gA,
                  const float* __restrict__ gC, const float* __restrict__ gQ,
                  const float* __restrict__ gR, const float* __restrict__ gm0,
                  const float* __restrict__ gP0, const float* __restrict__ gSA,
                  const float* __restrict__ gSC, float* __restrict__ muf,
                  float* __restrict__ Vf, int N) {
  __shared__ float B[15][ZD * MS];
  __shared__ float vec[6][ZD];
  float* sA  = B[0];  float* sAT = B[1];
  float* sCp = B[2];  float* sCpT = B[3];
  float* sSA = B[4];  float* sSC = B[5];
  float* sP  = B[6];  float* sL  = B[7];
  float* sLP = B[8];  float* sK  = B[9];
  float* sV  = B[10]; float* sG  = B[11];
  float* T1  = B[12]; float* T2  = B[13]; float* T3 = B[14];
  float* vm  = vec[0]; float* vmu = vec[1]; float* vLm = vec[2];
  float* vxh = vec[3]; float* vr  = vec[4]; float* vt = vec[5];
  const int l = threadIdx.x & 31;

  for (int i = 0; i < ZD; ++i) {
    sA[i * MS + l]   = gA[i * ZD + l];
    sAT[i * MS + l]  = gA[l * ZD + i];
    sCp[i * MS + l]  = (i < XDIM) ? gC[i * ZD + l] : 0.0f;   // C padded rows
    sCpT[i * MS + l] = (l < XDIM) ? gC[l * ZD + i] : 0.0f;
    sSA[i * MS + l]  = gSA[i * ZD + l];
    sSC[i * MS + l]  = gSC[i * ZD + l];
    sP[i * MS + l]   = gP0[i * ZD + l];
  }
  vm[l] = gm0[l];
  __syncthreads();

  for (int n = 0; n < N; ++n) {
    if (n + 8 < N)  // gfx1250 global_prefetch_b8 on the observation stream
      __builtin_prefetch(&gx[(size_t)(l & 15) * N + n + 8], 0, 0);

    // ---- update ----
    woodbury(sP, sSC, sL, T1, T2, T3);     // L
    mm32(sL, sP, sLP);                     // L P
    mm32(sLP, sCpT, T1);                   // L P C^T (padded)
    mm32(sCp, T1, T2);                     // C L P C^T
    for (int i = 0; i < ZD; ++i) {         // += [[R,0],[0,I]]
      float add = (i < XDIM && l < XDIM) ? gR[i * XDIM + l]
                  : ((i == l && i >= XDIM) ? 1.0f : 0.0f);
      T2[i * MS + l] += add;
    }
    __syncthreads();
    inv32(T2, T3);                         // inv(S) padded
    mm32(T1, T3, sK);                      // K (left 16 cols valid)
    mv32(sL, vm, vLm);                     // L m
    mv32(sCp, vLm, vxh);                   // x_hat
    vr[l] = ((l < XDIM) ? gx[(size_t)l * N + n] : 0.0f) - vxh[l];
    __syncthreads();
    mv32(sK, vr, vt);                      // K (x - x_hat)
    vmu[l] = vLm[l] + vt[l];
    __syncthreads();
    mm32(sK, sCp, T1);                     // K C
    negAddI(T1);                           // I - K C
    mm32(T1, sLP, sV);                     // V
    symM(sV, T2);

    muf[(size_t)l * N + n] = vmu[l];
    for (int i = 0; i < ZD; ++i)
      Vf[((size_t)(i * ZD + l)) * N + n] = sV[i * MS + l];

    // ---- predict ----
    woodbury(sV, sSA, sG, T1, T2, T3);     // G
    mv32(sG, vmu, vt);                     // G mu
    mv32(sA, vt, vm);                      // m_next
    mm32(sG, sV, T1);                      // G V
    mm32(sA, T1, T2);                      // A G V
    mm32(T2, sAT, sP);                     // A G V A^T
    for (int i = 0; i < ZD; ++i) sP[i * MS + l] += gQ[i * ZD + l];
    __syncthreads();
    symM(sP, T3);
  }
}

// ---------------------------------------------------------------------------
// Kernel 2: backward RTS smoother (sequential, reverse). One wave32.
// Writes mus, Vs (size N) and V12 (size N-1).
// ---------------------------------------------------------------------------
__global__ __launch_bounds__(32)
void vblds_smoother(const float* __restrict__ gA, const float* __restrict__ gQ,
                    const float* __restrict__ gSA, const float* __restrict__ muf,
                    const float* __restrict__ Vf, float* __restrict__ mus,
                    float* __restrict__ Vs, float* __restrict__ V12, int N) {
  __shared__ float B[14][ZD * MS];
  __shared__ float vec[6][ZD];
  float* sA  = B[0];  float* sAT = B[1];  float* sSA = B[2];
  float* sVf = B[3];  float* sVp = B[4];  float* sP  = B[5];
  float* sG  = B[6];  float* sGV = B[7];  float* sJ  = B[8];
  float* sJT = B[9];  float* sW  = B[10];
  float* T1  = B[11]; float* T2  = B[12]; float* T3  = B[13];
  float* vmf = vec[0]; float* vmp = vec[1]; float* vGm = vec[2];
  float* vm  = vec[3]; float* vdv = vec[4]; float* vt  = vec[5];
  const int l = threadIdx.x & 31;

  for (int i = 0; i < ZD; ++i) {
    sA[i * MS + l]  = gA[i * ZD + l];
    sAT[i * MS + l] = gA[l * ZD + i];
    sSA[i * MS + l] = gSA[i * ZD + l];
    sVp[i * MS + l] = Vf[((size_t)(i * ZD + l)) * N + (N - 1)];
  }
  vmp[l] = muf[(size_t)l * N + (N - 1)];
  __syncthreads();

  mus[(size_t)l * N + (N - 1)] = vmp[l];
  for (int i = 0; i < ZD; ++i)
    Vs[((size_t)(i * ZD + l)) * N + (N - 1)] = sVp[i * MS + l];

  for (int n = N - 2; n >= 0; --n) {
    for (int i = 0; i < ZD; ++i)
      sVf[i * MS + l] = Vf[((size_t)(i * ZD + l)) * N + n];
    vmf[l] = muf[(size_t)l * N + n];
    __syncthreads();

    // predict(mufn, Vfn)
    woodbury(sVf, sSA, sG, T1, T2, T3);    // G
    mv32(sG, vmf, vGm);                    // G mufn
    mv32(sA, vGm, vm);                     // m
    mm32(sG, sVf, sGV);                    // G Vfn
    mm32(sA, sGV, T1);
    mm32(T1, sAT, sP);                     // A G Vfn A^T
    for (int i = 0; i < ZD; ++i) sP[i * MS + l] += gQ[i * ZD + l];
    __syncthreads();
    symM(sP, T2);                          // P

    // J = G Vfn A^T inv(P)
    copyM(sP, T1);
    inv32(T1, T2);                         // inv(P)
    mm32(sGV, sAT, T3);
    mm32(T3, T2, sJ);
    transM(sJ, sJT);

    // mu = G mufn + J (mu_plus - m)
    vdv[l] = vmp[l] - vm[l];
    __syncthreads();
    mv32(sJ, vdv, vt);
    vt[l] = vGm[l] + vt[l];
    __syncthreads();

    // V = sym(G Vfn + J (Vp - P) J^T)
    for (int i = 0; i < ZD; ++i) T1[i * MS + l] = sVp[i * MS + l] - sP[i * MS + l];
    __syncthreads();
    mm32(sJ, T1, T3);
    mm32(T3, sJT, sW);
    for (int i = 0; i < ZD; ++i) sW[i * MS + l] += sGV[i * MS + l];
    __syncthreads();
    symM(sW, T1);

    mm32(sJ, sVp, T3);                     // V12 = J Vp

    mus[(size_t)l * N + n] = vt[l];
    for (int i = 0; i < ZD; ++i) {
      Vs[((size_t)(i * ZD + l)) * N + n]               = sW[i * MS + l];
      V12[((size_t)(i * ZD + l)) * (size_t)(N - 1) + n] = T3[i * MS + l];
    }
    vmp[l] = vt[l];
    __syncthreads();
    copyM(sW, sVp);
  }
}

// ---------------------------------------------------------------------------
// Kernel 3: bandwidth-bound reductions over n.  One block per Gram element.
// ws layout (floats): [0:1024) SVS  [1024:2048) SV12  [2048:3072) MUMU
//                     [3072:4096) MM12  [4096:4608) ZX  [4608:4864) XX
// ---------------------------------------------------------------------------
__global__ __launch_bounds__(256)
void vblds_reduce(const float* __restrict__ gx, const float* __restrict__ mus,
                  const float* __restrict__ Vs, const float* __restrict__ V12,
                  float* __restrict__ ws, int N) {
  const int e = blockIdx.x;  // 0..1023
  const int t = threadIdx.x;
  const int i = e >> 5, j = e & 31;
  float s_vs = 0, s_v12 = 0, s_mm = 0, s_mm12 = 0, s_zx = 0, s_xx = 0;
  const float* mui  = mus + (size_t)i * N;
  const float* muj  = mus + (size_t)j * N;
  const float* vse  = Vs + (size_t)e * N;
  const float* v12e = V12 + (size_t)e * (size_t)(N - 1);
  for (int n = t; n < N; n += 256) {
    float mi = mui[n], mj = muj[n];
    s_vs += vse[n];
    s_mm += mi * mj;
    if (n < N - 1) {
      s_v12  += v12e[n];
      s_mm12 += mi * muj[n + 1];
    }
  }
  if (e < 512) {
    const float* mu2 = mus + (size_t)(e >> 4) * N;
    const float* x2  = gx + (size_t)(e & 15) * N;
    for (int n = t; n < N; n += 256) s_zx += mu2[n] * x2[n];
  }
  if (e < 256) {
    const float* xa = gx + (size_t)(e >> 4) * N;
    const float* xb = gx + (size_t)(e & 15) * N;
    for (int n = t; n < N; n += 256) s_xx += xa[n] * xb[n];
  }
  __shared__ float red[256];
  auto blockReduce = [&](float v) -> float {
    red[t] = v;
    __syncthreads();
    for (int s = 128; s > 0; s >>= 1) {
      if (t < s) red[t] += red[t + s];
      __syncthreads();
    }
    float r = red[0];
    __syncthreads();
    return r;
  };
  float r0 = blockReduce(s_vs);
  float r1 = blockReduce(s_v12);
  float r2 = blockReduce(s_mm);
  float r3 = blockReduce(s_mm12);
  float r4 = blockReduce(s_zx);
  float r5 = blockReduce(s_xx);
  if (t == 0) {
    ws[e]        = r0;
    ws[1024 + e] = r1;
    ws[2048 + e] = r2;
    ws[3072 + e] = r3;
    if (e < 512) ws[4096 + e] = r4;
    if (e < 256) ws[4608 + e] = r5;
  }
}

// ---------------------------------------------------------------------------
// Kernel 4: M-step + log-likelihood.  One wave32, WMMA + GJ logdet.
// Uses exact trace simplifications: tr(inv(Qn)quadz)=(N-1)*32,
// tr(inv(Rn)quadx)=N*16, quad0=P0_new => tr=32.
// ---------------------------------------------------------------------------
__global__ __launch_bounds__(32)
void vblds_finalize(const float* __restrict__ ws, const float* __restrict__ mus,
                    const float* __restrict__ Vs, float* __restrict__ out, int N) {
  __shared__ float B[12][ZD * MS];
  __shared__ float mu0[ZD], muL[ZD];
  const int l = threadIdx.x & 31;
  float* svs = B[0];  float* sv12 = B[1]; float* smm = B[2]; float* smm12 = B[3];
  float* vfirst = B[4]; float* vlast = B[5];
  float* zz = B[6]; float* z22 = B[7];
  float* T1 = B[8]; float* T2 = B[9]; float* T3 = B[10]; float* T4 = B[11];

  for (int i = 0; i < ZD; ++i) {
    svs[i * MS + l]    = ws[i * ZD + l];
    sv12[i * MS + l]   = ws[1024 + i * ZD + l];
    smm[i * MS + l]    = ws[2048 + i * ZD + l];
    smm12[i * MS + l]  = ws[3072 + i * ZD + l];
    vfirst[i * MS + l] = Vs[((size_t)(i * ZD + l)) * N + 0];
    vlast[i * MS + l]  = Vs[((size_t)(i * ZD + l)) * N + (N - 1)];
  }
  mu0[l] = mus[(size_t)l * N + 0];
  muL[l] = mus[(size_t)l * N + (N - 1)];
  __syncthreads();
  const float LOG2PI = 1.8378770664093453f;

  for (int i = 0; i < ZD; ++i)
    zz[i * MS + l] = smm[i * MS + l] + svs[i * MS + l];
  for (int i = 0; i < ZD; ++i)
    z22[i * MS + l] = smm[i * MS + l] - mu0[i] * mu0[l] + svs[i * MS + l] - vfirst[i * MS + l];
  for (int i = 0; i < ZD; ++i)   // svs := z1z1
    svs[i * MS + l] = smm[i * MS + l] - muL[i] * muL[l] + svs[i * MS + l] - vlast[i * MS + l];
  for (int i = 0; i < ZD; ++i)   // sv12 := z1z2
    sv12[i * MS + l] += smm12[i * MS + l];
  __syncthreads();

  // ell_z0: quad0 == P0_new -> trace term = ZD
  symM(vfirst, T1);
  copyM(vfirst, T1);
  float ld0 = inv32(T1, T2);
  float ell_z0 = -0.5f * ((float)ZD * LOG2PI + ld0) - 0.5f * (float)ZD;

  // A_new = z1z2^T inv(z1z1)
  copyM(svs, T1);
  inv32(T1, T2);
  transM(sv12, T3);
  mm32(T3, T2, smm12);           // smm12 := A_new
  float* An = smm12;

  // quadz = sym(z2z2 - (An z12 + (An z12)^T) + An z11 An^T)
  mm32(An, sv12, T1);
  mm32(An, svs, T2);
  transM(An, T3);
  mm32(T2, T3, T4);
  for (int i = 0; i < ZD; ++i)
    smm[i * MS + l] = z22[i * MS + l] - T1[i * MS + l] - T1[l * MS + i] + T4[i * MS + l];
  __syncthreads();
  symM(smm, T1);
  copyM(smm, T1);
  float ldq = inv32(T1, T2);
  float logdetQ = ldq - (float)ZD * logf((float)(N - 1));
  float ell_z = -0.5f * (float)(N - 1) * ((float)ZD * LOG2PI + logdetQ)
                - 0.5f * (float)(N - 1) * (float)ZD;

  // x-side: C_new = zx^T inv(zz); quadx = xx - (Czx + Czx^T) + C zz C^T
  copyM(zz, T1);
  inv32(T1, T2);                 // inv(zz)
  for (int i = 0; i < ZD; ++i)   // zx^T padded (top 16 rows)
    T3[i * MS + l] = (i < XDIM) ? ws[4096 + l * XDIM + i] : 0.0f;
  __syncthreads();
  mm32(T3, T2, T4);              // T4 := C_new padded
  for (int i = 0; i < ZD; ++i)   // zx padded (left 16 cols)
    T1[i * MS + l] = (l < XDIM) ? ws[4096 + i * XDIM + l] : 0.0f;
  __syncthreads();
  mm32(T4, T1, T3);              // T3 := Czx (top-left 16x16)
  mm32(T4, zz, T1);              // Cn zz
  transM(T4, T2);                // Cn^T
  mm32(T1, T2, z22);             // z22 := CzzC
  for (int i = 0; i < ZD; ++i) {
    float q;
    if (i < XDIM && l < XDIM)
      q = ws[4608 + i * XDIM + l] - T3[i * MS + l] - T3[l * MS + i] + z22[i * MS + l];
    else
      q = (i == l) ? 1.0f : 0.0f;  // identity pad -> logdet unchanged
    zz[i * MS + l] = q;            // zz := quadx padded
  }
  __syncthreads();
  symM(zz, T1);
  copyM(zz, T1);
  float ldx = inv32(T1, T2);
  float logdetR = ldx - (float)XDIM * logf((float)N);
  float ell_x = -0.5f * (float)N * ((float)XDIM * LOG2PI + logdetR)
                - 0.5f * (float)N * (float)XDIM;

  if (l == 0) out[0] = ell_z + ell_x + ell_z0;
}

// ---------------------------------------------------------------------------
extern "C" void kernel_launch(void* const* d_in, const int* in_sizes, int n_in,
                              void* d_out, int out_size, void* d_ws, size_t ws_size,
                              hipStream_t stream) {
  const float* gx  = (const float*)d_in[0];
  const float* gA  = (const float*)d_in[1];
  const float* gC  = (const float*)d_in[2];
  const float* gQ  = (const float*)d_in[3];
  const float* gR  = (const float*)d_in[4];
  const float* gm0 = (const float*)d_in[5];
  const float* gP0 = (const float*)d_in[6];
  const float* gSA = (const float*)d_in[7];
  const float* gSC = (const float*)d_in[8];
  float* out = (float*)d_out;
  float* ws  = (float*)d_ws;
  const int N = in_sizes[0] / XDIM;

  // Output layout (floats): ell | muf(32N) | Vf(1024N) | mus(32N) | Vs(1024N)
  //                         | V12(1024(N-1))
  float* muf = out + 1;
  float* Vf  = muf + (size_t)32 * N;
  float* mus = Vf + (size_t)1024 * N;
  float* Vs  = mus + (size_t)32 * N;
  float* V12 = Vs + (size_t)1024 * N;

  vblds_filter<<<1, 32, 0, stream>>>(gx, gA, gC, gQ, gR, gm0, gP0, gSA, gSC,
                                     muf, Vf, N);
  vblds_smoother<<<1, 32, 0, stream>>>(gA, gQ, gSA, muf, Vf, mus, Vs, V12, N);
  vblds_reduce<<<1024, 256, 0, stream>>>(gx, mus, Vs, V12, ws, N);
  vblds_finalize<<<1, 32, 0, stream>>>(ws, mus, Vs, out, N);
}